// OnsiteModel_33913061769826
// MI455X (gfx1250) — compile-verified
//
#include <hip/hip_runtime.h>
#include <hip/hip_bf16.h>

// MI455X / gfx1250, wave32. All matrix math via V_WMMA_F32_16X16X4_F32
// (fp32 exact vs the fp32 reference). One block = 16 rows through the whole
// network; 16 waves each own 16x16 WMMA output tiles (12 scalar + 12 vector
// tiles per layer). Pair features are generated on the fly from LDS with
// branch-free split loops and packed u32 index-table reads.

typedef __attribute__((ext_vector_type(2))) float v2f;
typedef __attribute__((ext_vector_type(8))) float v8f;

#define THREADS 512
#define INV_SQRT3 0.57735026918962576f

struct Lds {
  float sA[16][129];              // scalar channels (padded: stride % 64 != 0)
  float vA[16][193];              // vector channels, [w*3+i] packed (padded)
  float gateB[16][65];            // sigmoid gates for the 64 vector channels
  unsigned short tab_ss2[8256];   // triu(128) pairs, u | (v<<8)  (4B aligned)
  unsigned short tab_dd2[2080];   // triu(64)
  unsigned short tab_ss1[528];    // triu(32)
  unsigned short tab_dd1[136];    // triu(16)
};

__device__ __forceinline__ v8f wmma_f32(v8f acc, v2f A, v2f B) {
  return __builtin_amdgcn_wmma_f32_16x16x4_f32(false, A, false, B, (short)0, acc,
                                               false, false);
}

// One 16x16 output tile of out_s[16 rows][16 cols] = ps[16][P0] * W0[m][P0]^T.
// K-space: [0, NSS) products s[u]*s[v]; [NSS, P0) dots (v[u].v[v])/sqrt(3).
// Both boundaries are multiples of 4 -> two branch-free loops.
template <int P0, int NSS>
__device__ __forceinline__ v8f gemm_scalar_tile(const float* __restrict__ W0, int ct,
                                                const unsigned short* tss,
                                                const unsigned short* tdd,
                                                const float* sRow, const float* vRow,
                                                int lane) {
  v8f acc = {};
  const int m = ct * 16 + (lane & 15);
  const int kPar = (lane >> 4) << 1;  // lanes 0-15 -> K=k,k+1 ; lanes 16-31 -> K=k+2,k+3
  const float* wp = W0 + (size_t)m * P0;
#pragma unroll 2
  for (int kb = 0; kb < NSS; kb += 4) {
    const int k0 = kb + kPar;
    const float2 bb = *(const float2*)(wp + k0);          // one global b64 per lane
    const unsigned tt = *(const unsigned*)(tss + k0);     // two packed index pairs
    v2f A, B;
    A.x = sRow[tt & 255u] * sRow[(tt >> 8) & 255u];
    A.y = sRow[(tt >> 16) & 255u] * sRow[tt >> 24];
    B.x = bb.x; B.y = bb.y;
    acc = wmma_f32(acc, A, B);
  }
#pragma unroll 2
  for (int kb = NSS; kb < P0; kb += 4) {
    const int k0 = kb + kPar;
    const float2 bb = *(const float2*)(wp + k0);
    const unsigned tt = *(const unsigned*)(tdd + (k0 - NSS));
    const float* a0 = vRow + 3 * (tt & 255u);
    const float* b0 = vRow + 3 * ((tt >> 8) & 255u);
    const float* a1 = vRow + 3 * ((tt >> 16) & 255u);
    const float* b1 = vRow + 3 * (tt >> 24);
    v2f A, B;
    A.x = (a0[0] * b0[0] + a0[1] * b0[1] + a0[2] * b0[2]) * INV_SQRT3;
    A.y = (a1[0] * b1[0] + a1[1] * b1[1] + a1[2] * b1[2]) * INV_SQRT3;
    B.x = bb.x; B.y = bb.y;
    acc = wmma_f32(acc, A, B);
  }
  return acc;
}

// One 16x16 tile of out_v[:, :, comp] = pv[16][N0*N1] * W1[m][N0*N1]^T with
// pv[u*N1+w] = s[u]*v[w][comp]. N1 % 4 == 0 -> each K=4 chunk has a single u,
// so s[u] is hoisted (one ds load per N1/4 WMMAs).
template <int N0, int N1>
__device__ __forceinline__ v8f gemm_vector_tile(const float* __restrict__ W1, int nt,
                                                int comp, const float* sRow,
                                                const float* vRow, int lane) {
  v8f acc = {};
  const int m = nt * 16 + (lane & 15);
  const int kPar = (lane >> 4) << 1;
  const float* wp = W1 + (size_t)m * (N0 * N1);
  for (int u = 0; u < N0; ++u) {
    const float su = sRow[u];
    const float* wrow = wp + u * N1;
#pragma unroll 4
    for (int wb = 0; wb < N1; wb += 4) {
      const int w0 = wb + kPar;
      const float2 bb = *(const float2*)(wrow + w0);
      v2f A, B;
      A.x = su * vRow[w0 * 3 + comp];
      A.y = su * vRow[w0 * 3 + 3 + comp];
      B.x = bb.x; B.y = bb.y;
      acc = wmma_f32(acc, A, B);
    }
  }
  return acc;
}

// One full TensorSquare + Gate layer on a 16-row tile resident in LDS.
// Tiles: t in [0,12) -> scalar GEMM column tile t; t in [12,24) -> vector tile
// (comp = (t-12)>>2, nt = (t-12)&3). Wave w owns t=w, and t=w+16 if w<8.
template <int N0, int N1>
__device__ void run_layer(const float* __restrict__ W0, const float* __restrict__ W1,
                          Lds& L, const unsigned short* tss,
                          const unsigned short* tdd) {
  constexpr int NSS = N0 * (N0 + 1) / 2;
  constexpr int NDD = N1 * (N1 + 1) / 2;
  constexpr int P0 = NSS + NDD;
  constexpr int P1 = N0 * N1;

  const int tid = threadIdx.x;
  const int wave = tid >> 5;
  const int lane = tid & 31;
  const int r = lane & 15;
  const float* sRow = L.sA[r];
  const float* vRow = L.vA[r];
  const float invS0 = 1.0f / sqrtf((float)P0);
  const float invS1 = 1.0f / sqrtf((float)P1);

  const int t0 = wave;
  const bool hasT1 = (wave < 8);

  v8f acc0 = {};
  v8f acc1 = {};
  if (t0 < 12) {
    acc0 = gemm_scalar_tile<P0, NSS>(W0, t0, tss, tdd, sRow, vRow, lane);
  } else {
    const int vt = t0 - 12;
    acc0 = gemm_vector_tile<N0, N1>(W1, vt & 3, vt >> 2, sRow, vRow, lane);
  }
  if (hasT1) {
    const int vt = wave + 4;  // t1 = wave + 16  ->  vt = wave + 4
    acc1 = gemm_vector_tile<N0, N1>(W1, vt & 3, vt >> 2, sRow, vRow, lane);
  }

  __syncthreads();  // all reads of old sA/vA complete

  const int rbase = (lane >> 4) << 3;  // C/D layout: VGPR g -> row g (+8 hi lanes)
  if (t0 < 12) {
    const int col = t0 * 16 + (lane & 15);
    if (t0 < 8) {  // silu region (cols 0..127) -> new scalar channels
#pragma unroll
      for (int g = 0; g < 8; ++g) {
        const float val = acc0[g] * invS0;
        L.sA[g + rbase][col] = val / (1.0f + __expf(-val));
      }
    } else {       // gate region (cols 128..191) -> sigmoid gates
#pragma unroll
      for (int g = 0; g < 8; ++g) {
        const float val = acc0[g] * invS0;
        L.gateB[g + rbase][col - 128] = 1.0f / (1.0f + __expf(-val));
      }
    }
  }

  __syncthreads();  // gates ready

  if (t0 >= 12) {
    const int vt = t0 - 12;
    const int mcol = (vt & 3) * 16 + (lane & 15);
    const int comp = vt >> 2;
#pragma unroll
    for (int g = 0; g < 8; ++g) {
      const int row = g + rbase;
      L.vA[row][mcol * 3 + comp] = acc0[g] * invS1 * L.gateB[row][mcol];
    }
  }
  if (hasT1) {
    const int vt = wave + 4;
    const int mcol = (vt & 3) * 16 + (lane & 15);
    const int comp = vt >> 2;
#pragma unroll
    for (int g = 0; g < 8; ++g) {
      const int row = g + rbase;
      L.vA[row][mcol * 3 + comp] = acc1[g] * invS1 * L.gateB[row][mcol];
    }
  }
  __syncthreads();  // new sA/vA ready
}

__global__ __launch_bounds__(THREADS) void onsite_tsq_kernel(
    const float* __restrict__ x, const float* __restrict__ W0_1,
    const float* __restrict__ W1_1, const float* __restrict__ W0_2,
    const float* __restrict__ W1_2, const float* __restrict__ W0_f,
    float* __restrict__ out) {
  __shared__ Lds L;
  const int tid = threadIdx.x;

  // Build triu index tables (u outer, v >= u: matches np.triu_indices order).
  if (tid < 128) {
    const int u = tid;
    int off = u * 128 - (u * (u - 1)) / 2;
    for (int v = u; v < 128; ++v) L.tab_ss2[off + v - u] = (unsigned short)(u | (v << 8));
  } else if (tid < 192) {
    const int u = tid - 128;
    int off = u * 64 - (u * (u - 1)) / 2;
    for (int v = u; v < 64; ++v) L.tab_dd2[off + v - u] = (unsigned short)(u | (v << 8));
  } else if (tid < 224) {
    const int u = tid - 192;
    int off = u * 32 - (u * (u - 1)) / 2;
    for (int v = u; v < 32; ++v) L.tab_ss1[off + v - u] = (unsigned short)(u | (v << 8));
  } else if (tid < 240) {
    const int u = tid - 224;
    int off = u * 16 - (u * (u - 1)) / 2;
    for (int v = u; v < 16; ++v) L.tab_dd1[off + v - u] = (unsigned short)(u | (v << 8));
  }

  // Load the 16-row input tile: x[:, :32] -> sA, x[:, 32:80] -> vA (w*3+i packed).
  const size_t rowBase = (size_t)blockIdx.x * 16;
  for (int idx = tid; idx < 16 * 80; idx += THREADS) {
    const int r = idx / 80, c = idx % 80;
    const float val = x[(rowBase + r) * 80 + c];
    if (c < 32) L.sA[r][c] = val;
    else        L.vA[r][c - 32] = val;
  }
  __syncthreads();

  // Layer 1: n0=32, n1=16 (P0=664, P1=512); Layer 2: n0=128, n1=64 (P0=10336, P1=8192)
  run_layer<32, 16>(W0_1, W1_1, L, L.tab_ss1, L.tab_dd1);
  run_layer<128, 64>(W0_2, W1_2, L, L.tab_ss2, L.tab_dd2);

  // Final: out[n] = ps2[n] . W0_f / sqrt(10336). One wave per row, split loops.
  const int wave = tid >> 5;
  const int lane = tid & 31;
  const float* sRow = L.sA[wave];
  const float* vRow = L.vA[wave];
  float part = 0.0f;
  for (int p = lane; p < 8256; p += 32) {
    const unsigned t = L.tab_ss2[p];
    part += sRow[t & 255u] * sRow[t >> 8] * W0_f[p];
  }
  for (int p = lane; p < 2080; p += 32) {
    const unsigned t = L.tab_dd2[p];
    const float* a = vRow + 3 * (t & 255u);
    const float* b = vRow + 3 * (t >> 8);
    part += (a[0] * b[0] + a[1] * b[1] + a[2] * b[2]) * INV_SQRT3 * W0_f[8256 + p];
  }
#pragma unroll
  for (int off = 16; off > 0; off >>= 1) part += __shfl_xor(part, off, 32);
  if (lane == 0) out[rowBase + wave] = part * (1.0f / sqrtf(10336.0f));
}

extern "C" void kernel_launch(void* const* d_in, const int* in_sizes, int n_in,
                              void* d_out, int out_size, void* d_ws, size_t ws_size,
                              hipStream_t stream) {
  const float* x    = (const float*)d_in[0];
  const float* W0_1 = (const float*)d_in[1];
  const float* W1_1 = (const float*)d_in[2];
  const float* W0_2 = (const float*)d_in[3];
  const float* W1_2 = (const float*)d_in[4];
  const float* W0_f = (const float*)d_in[5];
  float* out = (float*)d_out;

  const int N = in_sizes[0] / 80;   // rows (32768)
  dim3 grid(N / 16);
  onsite_tsq_kernel<<<grid, THREADS, 0, stream>>>(x, W0_1, W1_1, W0_2, W1_2, W0_f, out);
}